// Cost_volume_left_69818988364266
// MI455X (gfx1250) — compile-verified
//
#include <hip/hip_runtime.h>
#include <stdint.h>

typedef int v4i __attribute__((ext_vector_type(4)));
typedef int v8i __attribute__((ext_vector_type(8)));

#define Bd  8
#define Cd  64
#define Hd  256
#define Wd  256
#define Gd  8
#define CPG 8
#define ND  9

__global__ __launch_bounds__(256) void cost_volume_left_kernel(
    const float* __restrict__ featL,
    const float* __restrict__ featR,
    const float* __restrict__ disp,
    float* __restrict__ out)
{
    // One (b,h) row of feat_right, all 64 channels: 64 x 256 f32 = 64 KB LDS.
    __shared__ float sR[Cd * Wd];

    const int x = threadIdx.x;            // pixel column, 0..255
    const int b = blockIdx.x >> 8;        // blockIdx = b*H + h
    const int h = blockIdx.x & (Hd - 1);

    const size_t planeHW = (size_t)Hd * Wd;                       // 65536
    const size_t rowOff  = ((size_t)b * Cd) * planeHW + (size_t)h * Wd;

    // ---- Tensor Data Mover: DMA feat_right[b, :, h, :] into LDS (wave 0 only).
    if (x < 32) {
        const uint64_t gaddr = (uint64_t)(const void*)(featR + rowOff);
        const uint32_t laddr = (uint32_t)(size_t)(&sR[0]);  // low 32b of flat LDS addr = LDS offset
        v4i g0; v8i g1;
        // D# group 0: count=1 | lds_addr | global_addr[56:0] | type=2
        g0[0] = 1;
        g0[1] = (int)laddr;
        g0[2] = (int)(uint32_t)gaddr;
        g0[3] = (int)((uint32_t)(gaddr >> 32) | (2u << 30));
        // D# group 1: data_size=4B; tensor 256x64; tile 256x64; dim0_stride = H*W
        g1[0] = (2 << 16);                 // data_size = 2 -> 4 bytes; no mask/pad/iterate
        g1[1] = (Wd & 0xFFFF) << 16;       // tensor_dim0 = 256 (lo16)
        g1[2] = (Cd & 0xFFFF) << 16;       // tensor_dim0 hi=0 | tensor_dim1 = 64 (lo16)
        g1[3] = (Wd << 16);                // tensor_dim1 hi=0 | tile_dim0 = 256
        g1[4] = Cd;                        // tile_dim1 = 64, tile_dim2 = 0 (2D)
        g1[5] = (int)planeHW;              // tensor_dim0_stride lo32 = 65536 elements
        g1[6] = 0;                         // stride hi | dim1_stride lo (unused for 2D)
        g1[7] = 0;
        asm volatile("tensor_load_to_lds %0, %1" :: "s"(g0), "s"(g1) : "memory");
    }

    // ---- Per-pixel warp coefficients (overlapped with the TDM transfer).
    // xs_d = x - disp - off_d, off_d = d-4 integer  =>  frac is offset-independent.
    const float dval = disp[(size_t)b * planeHW + (size_t)h * Wd + x];
    const float t  = (float)x - dval;
    const float fF = floorf(t);
    const int   F  = (int)fF;
    const float w1 = t - fF;
    const float w0 = 1.0f - w1;

    // 10 clamped gather indices covering F-4 .. F+5 (all x0/x1 for the 9 offsets).
    int idxc[10];
    #pragma unroll
    for (int j = 0; j < 10; ++j) {
        int ix = F - 4 + j;
        idxc[j] = ix < 0 ? 0 : (ix > Wd - 1 ? Wd - 1 : ix);
    }
    // Zero-padding validity folded into per-offset weights.
    float w0m[ND], w1m[ND];
    #pragma unroll
    for (int d = 0; d < ND; ++d) {
        const int x0 = F + 4 - d;          // floor index for offset d-4
        const int x1 = x0 + 1;
        w0m[d] = (x0 >= 0 && x0 < Wd) ? w0 : 0.0f;
        w1m[d] = (x1 >= 0 && x1 < Wd) ? w1 : 0.0f;
    }

    if (x < 32) {
        __builtin_amdgcn_s_wait_tensorcnt(0);   // wave 0: DMA complete, LDS visible
    }
    __syncthreads();                            // rest of the workgroup ordered after it

    const size_t outBase = (size_t)b * (Gd * ND) * planeHW + (size_t)h * Wd + (size_t)x;
    const float* __restrict__ Lp = featL + rowOff + x;   // feat_left: stream, no reuse

    #pragma unroll 1
    for (int g = 0; g < Gd; ++g) {
        float P[10];
        #pragma unroll
        for (int j = 0; j < 10; ++j) P[j] = 0.0f;

        #pragma unroll
        for (int cc = 0; cc < CPG; ++cc) {
            const int c = g * CPG + cc;
            const float Lc = Lp[(size_t)c * planeHW];    // coalesced b32, imm offset
            const float* row = &sR[c * Wd];
            #pragma unroll
            for (int j = 0; j < 10; ++j) {
                P[j] = fmaf(Lc, row[idxc[j]], P[j]);     // shared partial sums
            }
        }
        #pragma unroll
        for (int d = 0; d < ND; ++d) {
            const float v = w0m[d] * P[8 - d] + w1m[d] * P[9 - d];
            out[outBase + (size_t)(g * ND + d) * planeHW] = 0.125f * v;  // mean over cpg=8
        }
    }
}

extern "C" void kernel_launch(void* const* d_in, const int* in_sizes, int n_in,
                              void* d_out, int out_size, void* d_ws, size_t ws_size,
                              hipStream_t stream) {
    (void)in_sizes; (void)n_in; (void)out_size; (void)d_ws; (void)ws_size;
    const float* featL = (const float*)d_in[0];
    const float* featR = (const float*)d_in[1];
    const float* dsp   = (const float*)d_in[2];
    float* out = (float*)d_out;

    dim3 grid(Bd * Hd);   // one workgroup per (batch, row)
    dim3 block(256);      // 8 wave32 waves
    hipLaunchKernelGGL(cost_volume_left_kernel, grid, block, 0, stream,
                       featL, featR, dsp, out);
}